// InteractionModule_76759655514800
// MI455X (gfx1250) — compile-verified
//
#include <hip/hip_runtime.h>
#include <hip/hip_bf16.h>
#include <math.h>

// ---------------------------------------------------------------------------
// MI455X / gfx1250, wave32. fp32 WMMA (V_WMMA_F32_16X16X4_F32) GEMM tiles.
// ---------------------------------------------------------------------------

typedef float v2f __attribute__((ext_vector_type(2)));
typedef float v8f __attribute__((ext_vector_type(8)));

#define EPS_F     1e-5f
#define INV_SQRT3 0.57735026918962576f
#define INV_SQRT2 0.70710678118654752f

__device__ __forceinline__ v8f wmma_f32_k4(v2f a, v2f b, v8f c) {
  // D = A(16x4 f32) * B(4x16 f32) + C(16x16 f32)
  return __builtin_amdgcn_wmma_f32_16x16x4_f32(
      /*neg_a=*/false, a, /*neg_b=*/false, b,
      /*c_mod=*/(short)0, c, /*reuse_a=*/false, /*reuse_b=*/false);
}

// A: 16 x K tile in LDS, row-major with row pitch lda.
// B: K x ldb matrix in global memory (row-major), columns [colBase, colBase+16).
// Per ISA layout: lane l holds A row (l&15), K pair offset ((l>>4)*2);
//                 B col (l&15), same K pair offset.  C/D: row r+(l>>4)*8, col l&15.
__device__ __forceinline__ v8f gemm_tile(const float* __restrict__ As, int lda,
                                         const float* __restrict__ Bg, int ldb,
                                         int colBase, int K, int lane) {
  const int mA = lane & 15;
  const int ko = (lane >> 4) << 1;
  const int col = colBase + (lane & 15);
  const float* __restrict__ arow = As + mA * lda + ko;
  const float* __restrict__ bcol = Bg + col + (long)ko * ldb;
  v8f c = {0.f, 0.f, 0.f, 0.f, 0.f, 0.f, 0.f, 0.f};
  for (int k = 0; k < K; k += 4) {
    v2f a, b;
    a.x = arow[k];
    a.y = arow[k + 1];
    b.x = bcol[(long)k * ldb];
    b.y = bcol[(long)k * ldb + ldb];
    c = wmma_f32_k4(a, b, c);
  }
  return c;
}

__device__ __forceinline__ void store_tile(float* __restrict__ Hs, int ldh,
                                           int colLoc, const float* __restrict__ bias,
                                           int colBias, bool do_silu, v8f c, int lane) {
  const int n = lane & 15;
  const int rb = (lane >> 4) << 3;
  const float bv = bias ? bias[colBias + n] : 0.f;
#pragma unroll
  for (int r = 0; r < 8; ++r) {
    float x = c[r] + bv;
    if (do_silu) x = x / (1.f + expf(-x));
    Hs[(rb + r) * ldh + colLoc + n] = x;
  }
}

// reduce-add across the 16-lane group (wave32: two groups per wave)
__device__ __forceinline__ float red16(float v) {
#pragma unroll
  for (int off = 8; off >= 1; off >>= 1) v += __shfl_xor(v, off, 16);
  return v;
}

__device__ __forceinline__ unsigned flip_f32(float f) {
  unsigned u = __float_as_uint(f);
  return (u & 0x80000000u) ? ~u : (u | 0x80000000u);
}
__device__ __forceinline__ float unflip_f32(unsigned u) {
  return (u & 0x80000000u) ? __uint_as_float(u & 0x7fffffffu) : __uint_as_float(~u);
}

// padded LDS pitches (pitch % 64 banks coprime-ish -> conflict-free column walks)
#define P16  18
#define P32  34
#define P48  50
#define P64  66
#define P128 130
#define P256 258

// ---------------------------------------------------------------------------
// Kernel 1: env MLP + gates, q/k projections, logits, env_s/env_v, seg-max.
// grid = E/16 blocks x 256 threads (8 waves).
// ---------------------------------------------------------------------------
__global__ __launch_bounds__(256) void k1_edge_env_attn(
    const float* __restrict__ node_inv, const float* __restrict__ slat,
    const float* __restrict__ eqlat, const int* __restrict__ ecenter,
    const float* __restrict__ w1, const float* __restrict__ b1,
    const float* __restrict__ w2, const float* __restrict__ b2,
    const float* __restrict__ w3, const float* __restrict__ b3,
    const float* __restrict__ wq, const float* __restrict__ wk,
    float* __restrict__ logits, float* __restrict__ envs,
    float* __restrict__ envv, unsigned* __restrict__ nmax) {
  __shared__ float Xs[16][P128];   // scalar_latent tile
  __shared__ float Ns[16][P64];    // gathered node invariants
  __shared__ float H1[16][P128];
  __shared__ float H2[16][P128];
  __shared__ float Gs[16][P32];    // gates
  __shared__ float Qm[16][P256];
  __shared__ float Km[16][P256];

  const int t = threadIdx.x;
  const int wave = t >> 5, lane = t & 31;
  const long eBase = (long)blockIdx.x * 16;

  // stage X tile (16 x 128), vectorized
  {
    const float4* __restrict__ src = (const float4*)(slat + eBase * 128);
    for (int i = t; i < 16 * 32; i += 256) {
      float4 v = src[i];
      int r = i >> 5, cc = (i & 31) << 2;
      Xs[r][cc] = v.x; Xs[r][cc + 1] = v.y; Xs[r][cc + 2] = v.z; Xs[r][cc + 3] = v.w;
    }
  }
  // stage gathered node invariants (16 x 64)
  {
    int r = t >> 4, j = (t & 15) << 2;
    int c = ecenter[eBase + r];
    float4 v = *(const float4*)(node_inv + (long)c * 64 + j);
    Ns[r][j] = v.x; Ns[r][j + 1] = v.y; Ns[r][j + 2] = v.z; Ns[r][j + 3] = v.w;
  }
  __syncthreads();

  {  // h1 = silu(X @ env_w1 + b1)
    v8f c = gemm_tile(&Xs[0][0], P128, w1, 128, wave * 16, 128, lane);
    store_tile(&H1[0][0], P128, wave * 16, b1, wave * 16, true, c, lane);
  }
  __syncthreads();
  {  // h2 = silu(h1 @ env_w2 + b2)
    v8f c = gemm_tile(&H1[0][0], P128, w2, 128, wave * 16, 128, lane);
    store_tile(&H2[0][0], P128, wave * 16, b2, wave * 16, true, c, lane);
  }
  __syncthreads();
  // tasks: [0,2) gate tiles, [2,18) k tiles, [18,34) q tiles
  for (int task = wave; task < 34; task += 8) {
    if (task < 2) {
      v8f c = gemm_tile(&H2[0][0], P128, w3, 32, task * 16, 128, lane);
      store_tile(&Gs[0][0], P32, task * 16, b3, task * 16, false, c, lane);
    } else if (task < 18) {
      int j = task - 2;
      v8f c = gemm_tile(&Xs[0][0], P128, wk, 256, j * 16, 128, lane);
      store_tile(&Km[0][0], P256, j * 16, nullptr, 0, false, c, lane);
    } else {
      int j = task - 18;
      v8f c = gemm_tile(&Ns[0][0], P64, wq, 256, j * 16, 64, lane);
      store_tile(&Qm[0][0], P256, j * 16, nullptr, 0, false, c, lane);
    }
  }
  __syncthreads();

  {  // per (edge, m): logits, env_s, env_v, seg-max
    const int el = t >> 4, m = t & 15;
    const long e = eBase + el;
    float lg = 0.f;
#pragma unroll
    for (int d = 0; d < 16; ++d) lg += Qm[el][m * 16 + d] * Km[el][m * 16 + d];
    lg *= 4.0f;  // reference: logits * isqrt(hd) == * 4
    float4 a = *(const float4*)(eqlat + (e * 16 + m) * 4);
    float gs = Gs[el][2 * m], gv = Gs[el][2 * m + 1];
    long em = e * 16 + m;
    logits[em] = lg;
    envs[em] = gs * a.x;
    envv[em * 3 + 0] = gv * a.y;
    envv[em * 3 + 1] = gv * a.z;
    envv[em * 3 + 2] = gv * a.w;
    int c = ecenter[e];
    atomicMax(&nmax[c * 16 + m], flip_f32(lg));
  }
}

// ---------------------------------------------------------------------------
// Kernel 2: ew = exp(logit - max); atomic segment sums into node buffers.
// ---------------------------------------------------------------------------
__global__ __launch_bounds__(256) void k2_scatter(
    const int* __restrict__ ecenter, const float* __restrict__ logits,
    const float* __restrict__ envs, const float* __restrict__ envv,
    const unsigned* __restrict__ nmax, float* __restrict__ nsum,
    float* __restrict__ ns, float* __restrict__ nv, long EM) {
  long idx = (long)blockIdx.x * 256 + threadIdx.x;
  if (idx >= EM) return;
  long e = idx >> 4;
  int m = (int)(idx & 15);
  int c = ecenter[e];
  float mx = unflip_f32(nmax[c * 16 + m]);
  float ew = expf(logits[idx] - mx);
  atomicAdd(&nsum[c * 16 + m], ew);
  atomicAdd(&ns[c * 16 + m], ew * envs[idx]);
  atomicAdd(&nv[(c * 16 + m) * 3 + 0], ew * envv[idx * 3 + 0]);
  atomicAdd(&nv[(c * 16 + m) * 3 + 1], ew * envv[idx * 3 + 1]);
  atomicAdd(&nv[(c * 16 + m) * 3 + 2], ew * envv[idx * 3 + 2]);
}

// ---------------------------------------------------------------------------
// Kernel 3: per-node softmax divide + irrep norms. 16 lanes per node.
// ---------------------------------------------------------------------------
__global__ __launch_bounds__(256) void k3_node_norm(
    const float* __restrict__ nsum, const float* __restrict__ ns,
    const float* __restrict__ nv, float* __restrict__ nsn,
    float* __restrict__ nvn, long NM) {
  long idx = (long)blockIdx.x * 256 + threadIdx.x;
  if (idx >= NM) return;
  float inv = 1.f / (nsum[idx] + 1e-12f);
  float s = ns[idx] * inv;
  float v0 = nv[idx * 3 + 0] * inv;
  float v1 = nv[idx * 3 + 1] * inv;
  float v2 = nv[idx * 3 + 2] * inv;
  float ssum = red16(s * s);
  float vsum = red16(v0 * v0 + v1 * v1 + v2 * v2);
  float si = 1.f / sqrtf(ssum * (1.f / 16.f) + EPS_F);
  float vi = 1.f / sqrtf(vsum * (1.f / 48.f) + EPS_F);
  nsn[idx] = s * si;
  nvn[idx * 3 + 0] = v0 * vi;
  nvn[idx * 3 + 1] = v1 * vi;
  nvn[idx * 3 + 2] = v2 * vi;
}

// ---------------------------------------------------------------------------
// Kernel 4: tensor products + norms, sc MLP + layernorm + residual, equiv out.
// grid = E/16 blocks x 256 threads.
// ---------------------------------------------------------------------------
__global__ __launch_bounds__(256) void k4_edge_out(
    const float* __restrict__ slat, const float* __restrict__ eqlat,
    const int* __restrict__ ecenter, const int* __restrict__ active,
    const float* __restrict__ ucoef,
    const float* __restrict__ w1, const float* __restrict__ b1,
    const float* __restrict__ w2, const float* __restrict__ b2,
    const float* __restrict__ w3, const float* __restrict__ b3,
    const float* __restrict__ lng, const float* __restrict__ lnb,
    const float* __restrict__ weqs, const float* __restrict__ weqv,
    const float* __restrict__ nsn, const float* __restrict__ nvn,
    float* __restrict__ outS, float* __restrict__ outE) {
  __shared__ float Si[16][P32];       // interleaved scalars (stack order) for MLP
  __shared__ float Sc[16][P32];       // concatenated scalars for eq_s
  __shared__ float Vs[3][16][P48];    // [component][edge][48] for eq_v
  __shared__ float H1[16][P128];
  __shared__ float H2[16][P128];
  __shared__ float NL[16][P128];
  __shared__ float Es[16][P16];
  __shared__ float Ev[3][16][P16];

  const int t = threadIdx.x, wave = t >> 5, lane = t & 31;
  const long eBase = (long)blockIdx.x * 16;
  const int el = t >> 4, m = t & 15;
  const long e = eBase + el;

  {  // tensor products + per-edge irrep norms
    int c = ecenter[e];
    float4 a = *(const float4*)(eqlat + (e * 16 + m) * 4);
    float bs = nsn[c * 16 + m];
    float b0 = nvn[(c * 16 + m) * 3 + 0];
    float b1v = nvn[(c * 16 + m) * 3 + 1];
    float b2v = nvn[(c * 16 + m) * 3 + 2];
    float s1 = a.x * bs;
    float s2 = (a.y * b0 + a.z * b1v + a.w * b2v) * INV_SQRT3;
    float v10 = a.x * b0, v11 = a.x * b1v, v12 = a.x * b2v;
    float v20 = a.y * bs, v21 = a.z * bs, v22 = a.w * bs;
    float v30 = (a.z * b2v - a.w * b1v) * INV_SQRT2;
    float v31 = (a.w * b0 - a.y * b2v) * INV_SQRT2;
    float v32 = (a.y * b1v - a.z * b0) * INV_SQRT2;
    float i1 = 1.f / sqrtf(red16(s1 * s1) * (1.f / 16.f) + EPS_F);
    float i2 = 1.f / sqrtf(red16(s2 * s2) * (1.f / 16.f) + EPS_F);
    float j1 = 1.f / sqrtf(red16(v10 * v10 + v11 * v11 + v12 * v12) * (1.f / 48.f) + EPS_F);
    float j2 = 1.f / sqrtf(red16(v20 * v20 + v21 * v21 + v22 * v22) * (1.f / 48.f) + EPS_F);
    float j3 = 1.f / sqrtf(red16(v30 * v30 + v31 * v31 + v32 * v32) * (1.f / 48.f) + EPS_F);
    Si[el][2 * m] = s1 * i1;  Si[el][2 * m + 1] = s2 * i2;
    Sc[el][m] = s1 * i1;      Sc[el][16 + m] = s2 * i2;
    Vs[0][el][m] = v10 * j1;  Vs[0][el][16 + m] = v20 * j2;  Vs[0][el][32 + m] = v30 * j3;
    Vs[1][el][m] = v11 * j1;  Vs[1][el][16 + m] = v21 * j2;  Vs[1][el][32 + m] = v31 * j3;
    Vs[2][el][m] = v12 * j1;  Vs[2][el][16 + m] = v22 * j2;  Vs[2][el][32 + m] = v32 * j3;
  }
  __syncthreads();
  {  // h = silu(scalars @ sc_w1 + b1), K=32
    v8f c = gemm_tile(&Si[0][0], P32, w1, 128, wave * 16, 32, lane);
    store_tile(&H1[0][0], P128, wave * 16, b1, wave * 16, true, c, lane);
  }
  __syncthreads();
  {
    v8f c = gemm_tile(&H1[0][0], P128, w2, 128, wave * 16, 128, lane);
    store_tile(&H2[0][0], P128, wave * 16, b2, wave * 16, true, c, lane);
  }
  __syncthreads();
  {
    v8f c = gemm_tile(&H2[0][0], P128, w3, 128, wave * 16, 128, lane);
    store_tile(&NL[0][0], P128, wave * 16, b3, wave * 16, false, c, lane);
  }
  __syncthreads();

  // waves 0..3: equivariant output GEMMs; all waves: layernorm + scalar_out
  if (wave == 0) {
    v8f c = gemm_tile(&Sc[0][0], P32, weqs, 16, 0, 32, lane);
    store_tile(&Es[0][0], P16, 0, nullptr, 0, false, c, lane);
  } else if (wave <= 3) {
    int d = wave - 1;
    v8f c = gemm_tile(&Vs[d][0][0], P48, weqv, 16, 0, 48, lane);
    store_tile(&Ev[d][0][0], P16, 0, nullptr, 0, false, c, lane);
  }
  {
    float uc = ucoef[0];
    float co = 1.f / sqrtf(uc * uc + 1.f);
    float cn = uc * co;
    float sm = 0.f, sq = 0.f;
#pragma unroll
    for (int j = 0; j < 8; ++j) {
      float x = NL[el][j * 16 + m];
      sm += x; sq += x * x;
    }
    sm = red16(sm); sq = red16(sq);
    float mean = sm * (1.f / 128.f);
    float var = sq * (1.f / 128.f) - mean * mean;
    float inv = 1.f / sqrtf(var + EPS_F);
    long orow = active[e];  // active_edges == arange(E): one update per row
#pragma unroll
    for (int j = 0; j < 8; ++j) {
      int col = j * 16 + m;
      float y = (NL[el][col] - mean) * inv * lng[col] + lnb[col];
      outS[orow * 128 + col] = co * slat[orow * 128 + col] + cn * y;
    }
  }
  __syncthreads();
  {  // equiv_out: (E, 16, 4) = [eq_s, eq_v]
    float4 o;
    o.x = Es[el][m];
    o.y = Ev[0][el][m];
    o.z = Ev[1][el][m];
    o.w = Ev[2][el][m];
    *(float4*)(outE + (e * 16 + m) * 4) = o;
  }
}

// ---------------------------------------------------------------------------
extern "C" void kernel_launch(void* const* d_in, const int* in_sizes, int n_in,
                              void* d_out, int out_size, void* d_ws, size_t ws_size,
                              hipStream_t stream) {
  const float* node_inv = (const float*)d_in[0];   // (N, 64)
  const float* slat     = (const float*)d_in[1];   // (E, 128)
  const float* eqlat    = (const float*)d_in[2];   // (E, 16, 4)
  const int*   eidx     = (const int*)d_in[3];     // (2, E)
  const int*   active   = (const int*)d_in[4];     // (E,)
  const float* ucoef    = (const float*)d_in[5];   // (1,)
  const float* env_w1 = (const float*)d_in[6];
  const float* env_b1 = (const float*)d_in[7];
  const float* env_w2 = (const float*)d_in[8];
  const float* env_b2 = (const float*)d_in[9];
  const float* env_w3 = (const float*)d_in[10];
  const float* env_b3 = (const float*)d_in[11];
  const float* wq     = (const float*)d_in[12];
  const float* wk     = (const float*)d_in[13];
  const float* sc_w1  = (const float*)d_in[14];
  const float* sc_b1  = (const float*)d_in[15];
  const float* sc_w2  = (const float*)d_in[16];
  const float* sc_b2  = (const float*)d_in[17];
  const float* sc_w3  = (const float*)d_in[18];
  const float* sc_b3  = (const float*)d_in[19];
  const float* ln_g   = (const float*)d_in[20];
  const float* ln_b   = (const float*)d_in[21];
  const float* weq_s  = (const float*)d_in[22];
  const float* weq_v  = (const float*)d_in[23];

  const long N = in_sizes[0] / 64;
  const long E = (long)in_sizes[1] / 128;
  const int* ecenter = eidx;  // edge_index[0]

  // workspace layout (floats)
  float* ws = (float*)d_ws;
  float*    logits = ws;                               // E*16
  float*    envs   = logits + E * 16;                  // E*16
  float*    envv   = envs + E * 16;                    // E*48
  unsigned* nmax   = (unsigned*)(envv + E * 48);       // N*16
  float*    nsum   = (float*)nmax + N * 16;            // N*16
  float*    ns     = nsum + N * 16;                    // N*16
  float*    nv     = ns + N * 16;                      // N*48
  float*    nsn    = nv + N * 48;                      // N*16
  float*    nvn    = nsn + N * 16;                     // N*48

  float* outS = (float*)d_out;           // (E, 128)
  float* outE = outS + E * 128;          // (E, 16, 4)

  // zero node accumulators (nmax flipped-uint 0 == below flip(-inf))
  hipMemsetAsync(nmax, 0, (size_t)N * (16 + 16 + 16 + 48) * sizeof(float), stream);

  dim3 blk(256);
  k1_edge_env_attn<<<dim3((unsigned)(E / 16)), blk, 0, stream>>>(
      node_inv, slat, eqlat, ecenter, env_w1, env_b1, env_w2, env_b2, env_w3,
      env_b3, wq, wk, logits, envs, envv, nmax);
  k2_scatter<<<dim3((unsigned)((E * 16 + 255) / 256)), blk, 0, stream>>>(
      ecenter, logits, envs, envv, nmax, nsum, ns, nv, E * 16);
  k3_node_norm<<<dim3((unsigned)((N * 16 + 255) / 256)), blk, 0, stream>>>(
      nsum, ns, nv, nsn, nvn, N * 16);
  k4_edge_out<<<dim3((unsigned)(E / 16)), blk, 0, stream>>>(
      slat, eqlat, ecenter, active, ucoef, sc_w1, sc_b1, sc_w2, sc_b2, sc_w3,
      sc_b3, ln_g, ln_b, weq_s, weq_v, nsn, nvn, outS, outE);
}